// BiMambaForMaskedLM_11940009082910
// MI455X (gfx1250) — compile-verified
//
#include <hip/hip_runtime.h>
#include <hip/hip_bf16.h>
#include <math.h>

// Model constants (from reference): L, DM, DI, DS, DR, KC, V, B, S
#define L_   4
#define DM_  768
#define DI_  1536
#define DS_  16
#define DR_  48
#define KC_  4
#define V_   32000
#define B_   2
#define S_   2048
#define NBS  (B_ * S_)          // 4096 token rows
#define DBC_ (DR_ + 2 * DS_)    // 80

#define GEMM_WAVES 8            // waves per block
#define GEMM_NT    4            // 16-wide N tiles per wave
#define GEMM_MT    2            // 16-row M tiles per block (shared A strips)
#define GEMM_KC    32           // K chunk staged in LDS

// Flip to 0 if the gfx1250 assembler rejects the async mnemonics.
#define USE_ASYNC_COPY 1

typedef __attribute__((ext_vector_type(2))) float v2f;
typedef __attribute__((ext_vector_type(8))) float v8f;

// ---------------------------------------------------------------------------
// fp32 WMMA GEMM:  C[M,N] = act( A[M,K] @ W[N,K]^T + bias )  (+= optional)
//
// Block = 256 threads = 8 waves, computes a 32-row M strip x (8*4) N tiles.
// Two 16x32 A strips are staged into double-buffered LDS with
// global_load_async_to_lds_b64 (ASYNCcnt): the copy for chunk i+1 is issued
// BEFORE computing chunk i and waited only at the end -> copy/compute overlap.
// Each wave holds 2x4 accumulators: one B fragment feeds 2 WMMAs, one A
// fragment feeds 4, giving 64 WMMAs per 32-wide K chunk.
// A-frag (32b 16x4): lanes 0-15 -> K=k..k+1 ; lanes 16-31 -> K=k+2..k+3
// B-frag (4x16):     lanes hold N=lane&15; VGPR0/1 hold K=2*half..2*half+1
// C/D (16x16 f32):   VGPR r: lanes 0-15 -> M=r ; lanes 16-31 -> M=r+8
// Partial K chunks (K=48 case) use a zero-padded synchronous path.
// N-tile clamping keeps every wave alive for the barriers; M % 32 == 0 always.
// ---------------------------------------------------------------------------
__global__ __launch_bounds__(256)
void wmma_gemm_f32_kernel(const float* __restrict__ A, long lda,
                          const float* __restrict__ W, long ldw,
                          const float* __restrict__ bias,
                          float* __restrict__ C, long ldc,
                          int M, int N, int K, int accumulate, int act)
{
  __shared__ float sA[2][GEMM_MT][16][34];  // double buffer; padded rows

  const int tiles_n = N >> 4;
  const int m0      = blockIdx.y * (16 * GEMM_MT);
  const int tn_base = blockIdx.x * (GEMM_WAVES * GEMM_NT);

  const int wave = threadIdx.x >> 5;
  const int lane = threadIdx.x & 31;
  const int half = lane >> 4;
  const int l15  = lane & 15;

  const float* wrow[GEMM_NT];
  int  tnj[GEMM_NT];
  bool keep[GEMM_NT];
#pragma unroll
  for (int j = 0; j < GEMM_NT; ++j) {
    int tn = tn_base + wave * GEMM_NT + j;
    keep[j] = (tn < tiles_n);            // wave-uniform
    if (!keep[j]) tn = tiles_n - 1;      // clamp: stay alive for barriers
    tnj[j]  = tn;
    wrow[j] = W + (long)(tn * 16 + l15) * ldw;
  }

  // Stage both 16x32 A strips for chunk ci into LDS buffer `buf`.
  auto stage = [&](int ci, int buf) {
    const int kbase = ci * GEMM_KC;
    const int kc = (K - kbase < GEMM_KC) ? (K - kbase) : GEMM_KC;
    const int t  = threadIdx.x;
    const int m  = t >> 4;               // 0..15
    const int kx = (t << 1) & 31;        // 0,2,...,30
#pragma unroll
    for (int mt = 0; mt < GEMM_MT; ++mt) {
      const float* gsrc = A + (long)(m0 + mt * 16 + m) * lda + kbase + kx;
#if USE_ASYNC_COPY
      if (kc == GEMM_KC) {
        // LDS offset = low 32 bits of the generic address of a __shared__ object
        unsigned ldsoff = (unsigned)(size_t)(void*)&sA[buf][mt][m][kx];
        asm volatile("global_load_async_to_lds_b64 %0, %1, off"
                     :: "v"(ldsoff), "v"(gsrc) : "memory");
      } else
#endif
      {
        float x0 = (kx     < kc) ? gsrc[0] : 0.0f;
        float x1 = (kx + 1 < kc) ? gsrc[1] : 0.0f;
        sA[buf][mt][m][kx]     = x0;
        sA[buf][mt][m][kx + 1] = x1;
      }
    }
  };

  v8f acc[GEMM_MT][GEMM_NT] = {};

  const int nchunks = (K + GEMM_KC - 1) / GEMM_KC;

  stage(0, 0);
#if USE_ASYNC_COPY
  asm volatile("s_wait_asynccnt 0x0" ::: "memory");
#endif
  __syncthreads();

  for (int ci = 0; ci < nchunks; ++ci) {
    const int buf = ci & 1;
    // kick off the next chunk's copy before computing this one
    if (ci + 1 < nchunks) stage(ci + 1, buf ^ 1);

    const int k0 = ci * GEMM_KC;
    const float* p[GEMM_NT];
#pragma unroll
    for (int j = 0; j < GEMM_NT; ++j) p[j] = wrow[j] + k0 + 2 * half;

    if (k0 + GEMM_KC < K) {
#pragma unroll
      for (int j = 0; j < GEMM_NT; ++j)
        __builtin_prefetch(&wrow[j][k0 + GEMM_KC], 0, 1);
    }

    // ---- 8 K-steps: batched B loads, then 2x4 WMMAs per step ----
#pragma unroll
    for (int kk = 0; kk < GEMM_KC; kk += 4) {
      const int k2 = kk + 2 * half;
      v2f a0, a1;
      a0[0] = sA[buf][0][l15][k2];  a0[1] = sA[buf][0][l15][k2 + 1];
      a1[0] = sA[buf][1][l15][k2];  a1[1] = sA[buf][1][l15][k2 + 1];
      v2f bf[GEMM_NT];
#pragma unroll
      for (int j = 0; j < GEMM_NT; ++j) { bf[j][0] = p[j][kk]; bf[j][1] = p[j][kk + 1]; }
#pragma unroll
      for (int j = 0; j < GEMM_NT; ++j) {
        acc[0][j] = __builtin_amdgcn_wmma_f32_16x16x4_f32(false, a0, false, bf[j],
                                                          (short)0, acc[0][j],
                                                          false, false);
        acc[1][j] = __builtin_amdgcn_wmma_f32_16x16x4_f32(false, a1, false, bf[j],
                                                          (short)0, acc[1][j],
                                                          false, false);
      }
    }

#if USE_ASYNC_COPY
    asm volatile("s_wait_asynccnt 0x0" ::: "memory");
#endif
    __syncthreads();
  }

  // ---- epilogue: bias / softplus / residual-accumulate, per kept tile ----
#pragma unroll
  for (int j = 0; j < GEMM_NT; ++j) {
    if (!keep[j]) continue;              // wave-uniform
    const int col = tnj[j] * 16 + l15;
    const float bv = bias ? bias[col] : 0.0f;
#pragma unroll
    for (int mt = 0; mt < GEMM_MT; ++mt) {
#pragma unroll
      for (int r = 0; r < 8; ++r) {
        const int mrow = m0 + mt * 16 + r + 8 * half;
        float v = acc[mt][j][r] + bv;
        if (act == 1) v = (v > 20.0f) ? v : __logf(1.0f + __expf(v)); // softplus
        const long idx = (long)mrow * ldc + col;
        if (accumulate) C[idx] += v; else C[idx] = v;
      }
    }
  }
}

// ---------------------------------------------------------------------------
// Embedding gather (with optional sequence flip for the backward direction)
// ---------------------------------------------------------------------------
__global__ __launch_bounds__(256)
void embed_kernel(const int* __restrict__ ids, const float* __restrict__ emb,
                  float* __restrict__ h, int flip)
{
  long i = (long)blockIdx.x * blockDim.x + threadIdx.x;
  const long total = (long)NBS * DM_;
  if (i >= total) return;
  int  d  = (int)(i % DM_);
  long bs = i / DM_;
  int  s  = (int)(bs % S_);
  long b  = bs / S_;
  int  ss = flip ? (S_ - 1 - s) : s;
  int  tok = ids[b * S_ + ss];
  h[i] = emb[(long)tok * DM_ + d];
}

// ---------------------------------------------------------------------------
// RMSNorm; strided/offset destination (for concat) and optional seq flip.
// ---------------------------------------------------------------------------
__global__ __launch_bounds__(256)
void rmsnorm_kernel(const float* __restrict__ x, const float* __restrict__ w,
                    float* __restrict__ out, int D, long out_stride, long out_off,
                    int flip)
{
  __shared__ float red[256];
  int row = blockIdx.x;             // over B*S
  const float* xr = x + (long)row * D;
  float ss = 0.0f;
  for (int i = threadIdx.x; i < D; i += 256) { float v = xr[i]; ss += v * v; }
  red[threadIdx.x] = ss;
  __syncthreads();
  for (int t = 128; t > 0; t >>= 1) {
    if (threadIdx.x < t) red[threadIdx.x] += red[threadIdx.x + t];
    __syncthreads();
  }
  float scale = rsqrtf(red[0] / (float)D + 1e-5f);
  int orow = row;
  if (flip) { int b = row / S_, t = row % S_; orow = b * S_ + (S_ - 1 - t); }
  float* o = out + (long)orow * out_stride + out_off;
  for (int i = threadIdx.x; i < D; i += 256) o[i] = xr[i] * scale * w[i];
}

// ---------------------------------------------------------------------------
// Causal depthwise conv (KC=4) + bias + SiLU on xi = xz[..., :DI]
// ---------------------------------------------------------------------------
__global__ __launch_bounds__(256)
void conv_silu_kernel(const float* __restrict__ xz, const float* __restrict__ convw,
                      const float* __restrict__ convb, float* __restrict__ xc)
{
  long i = (long)blockIdx.x * blockDim.x + threadIdx.x;
  const long total = (long)NBS * DI_;
  if (i >= total) return;
  int  d  = (int)(i % DI_);
  long bs = i / DI_;
  int  s  = (int)(bs % S_);
  long brow = bs - s;               // b*S
  float acc = convb[d];
#pragma unroll
  for (int k = 0; k < KC_; ++k) {
    int sp = s + k - (KC_ - 1);
    if (sp >= 0) acc += xz[(brow + sp) * (2L * DI_) + d] * convw[d * KC_ + k];
  }
  float sig = 1.0f / (1.0f + __expf(-acc));
  xc[i] = acc * sig;
}

// ---------------------------------------------------------------------------
// Sequential SSM scan over S. One thread per (b,d) channel, DS=16 states in
// registers. Per-timestep B/C staged in LDS. Fused +Dp*x and SiLU(z) gate.
// ---------------------------------------------------------------------------
__global__ __launch_bounds__(256)
void scan_kernel(const float* __restrict__ dt, const float* __restrict__ dbc,
                 const float* __restrict__ xc, const float* __restrict__ xz,
                 const float* __restrict__ Alog, const float* __restrict__ Dp,
                 float* __restrict__ y)
{
  __shared__ float sB[DS_], sC[DS_];
  const int blocksPerB = DI_ / 256;
  int b = blockIdx.x / blocksPerB;
  int d = (blockIdx.x % blocksPerB) * 256 + threadIdx.x;

  float Acoef[DS_], hstate[DS_];
#pragma unroll
  for (int n = 0; n < DS_; ++n) {
    Acoef[n]  = -__expf(Alog[(long)d * DS_ + n]);
    hstate[n] = 0.0f;
  }
  float Dpv = Dp[d];

  long row = (long)b * S_;
  for (int s = 0; s < S_; ++s, ++row) {
    if (threadIdx.x < 2 * DS_) {
      float v = dbc[row * DBC_ + DR_ + threadIdx.x];
      if (threadIdx.x < DS_) sB[threadIdx.x] = v; else sC[threadIdx.x - DS_] = v;
    }
    __syncthreads();
    float dtv = dt[row * DI_ + d];
    float xv  = xc[row * DI_ + d];
    float dx  = dtv * xv;
    float yacc = 0.0f;
#pragma unroll
    for (int n = 0; n < DS_; ++n) {
      hstate[n] = __expf(dtv * Acoef[n]) * hstate[n] + dx * sB[n];
      yacc += hstate[n] * sC[n];
    }
    float zv = xz[row * (2L * DI_) + DI_ + d];
    float sz = zv / (1.0f + __expf(-zv));
    y[row * DI_ + d] = (yacc + Dpv * xv) * sz;
    __syncthreads();
  }
}

// ---------------------------------------------------------------------------
// Per-row NLL via single-pass online logsumexp (halves logits re-read traffic)
// ---------------------------------------------------------------------------
__global__ __launch_bounds__(256)
void loss_row_kernel(const float* __restrict__ logits, const int* __restrict__ labels,
                     float* __restrict__ nll_mask)
{
  __shared__ float rm[256], rsum[256];
  int row = blockIdx.x;
  const float* lr = logits + (long)row * V_;

  float m = -3.402823466e38f, s = 0.0f;
  for (int i = threadIdx.x; i < V_; i += 256) {
    float v = lr[i];
    if (v > m) { s = s * __expf(m - v) + 1.0f; m = v; }
    else       { s += __expf(v - m); }
  }
  rm[threadIdx.x] = m; rsum[threadIdx.x] = s;
  __syncthreads();
  for (int t = 128; t > 0; t >>= 1) {
    if (threadIdx.x < t) {
      float ma = rm[threadIdx.x],  sa = rsum[threadIdx.x];
      float mb = rm[threadIdx.x + t], sb = rsum[threadIdx.x + t];
      float M  = fmaxf(ma, mb);
      rm[threadIdx.x]   = M;
      rsum[threadIdx.x] = sa * __expf(ma - M) + sb * __expf(mb - M);
    }
    __syncthreads();
  }
  if (threadIdx.x == 0) {
    int lab  = labels[row];
    int mask = (lab != -100);
    float lse = rm[0] + __logf(rsum[0]);
    nll_mask[row]       = mask ? (lse - lr[lab]) : 0.0f;
    nll_mask[NBS + row] = mask ? 1.0f : 0.0f;
  }
}

__global__ __launch_bounds__(256)
void loss_reduce_kernel(const float* __restrict__ nll_mask, float* __restrict__ out)
{
  __shared__ float rs[256], rm[256];
  float s = 0.0f, c = 0.0f;
  for (int i = threadIdx.x; i < NBS; i += 256) { s += nll_mask[i]; c += nll_mask[NBS + i]; }
  rs[threadIdx.x] = s; rm[threadIdx.x] = c;
  __syncthreads();
  for (int t = 128; t > 0; t >>= 1) {
    if (threadIdx.x < t) { rs[threadIdx.x] += rs[threadIdx.x + t]; rm[threadIdx.x] += rm[threadIdx.x + t]; }
    __syncthreads();
  }
  if (threadIdx.x == 0) out[0] = rs[0] / fmaxf(rm[0], 1.0f);
}

// ---------------------------------------------------------------------------
static inline void launch_gemm(const float* A, long lda, const float* W, long ldw,
                               const float* bias, float* C, long ldc,
                               int M, int N, int K, int accum, int act,
                               hipStream_t stream)
{
  int tiles_n = N / 16;
  dim3 blk(256);
  dim3 grd((unsigned)((tiles_n + GEMM_WAVES * GEMM_NT - 1) / (GEMM_WAVES * GEMM_NT)),
           (unsigned)(M / (16 * GEMM_MT)));
  wmma_gemm_f32_kernel<<<grd, blk, 0, stream>>>(A, lda, W, ldw, bias, C, ldc,
                                                M, N, K, accum, act);
}

extern "C" void kernel_launch(void* const* d_in, const int* in_sizes, int n_in,
                              void* d_out, int out_size, void* d_ws, size_t ws_size,
                              hipStream_t stream)
{
  (void)in_sizes; (void)n_in; (void)out_size; (void)ws_size;

  const int*   ids    = (const int*)d_in[0];
  const int*   labels = (const int*)d_in[1];
  const float* lm_w   = (const float*)d_in[26];
  float*       logits = (float*)d_out;               // (B,S,V) then loss scalar

  // Workspace partition (floats): ~178 MB total
  float* ws   = (float*)d_ws;
  float* h    = ws;                                   // NBS*DM
  float* xn   = h    + (long)NBS * DM_;               // NBS*DM
  float* xz   = xn   + (long)NBS * DM_;               // NBS*2DI
  float* xc   = xz   + (long)NBS * 2 * DI_;           // NBS*DI
  float* dbc  = xc   + (long)NBS * DI_;               // NBS*80
  float* dt   = dbc  + (long)NBS * DBC_;              // NBS*DI
  float* yb   = dt   + (long)NBS * DI_;               // NBS*DI
  float* comb = yb   + (long)NBS * DI_;               // NBS*2DM (concat dest)
  float* nll  = comb + (long)NBS * 2 * DM_;           // 2*NBS

  const long embTot  = (long)NBS * DM_;
  const long convTot = (long)NBS * DI_;

  for (int dir = 0; dir < 2; ++dir) {
    int base = 2 + dir * 12;
    const float* emb   = (const float*)d_in[base + 0];
    const float* norm  = (const float*)d_in[base + 1];
    const float* inw   = (const float*)d_in[base + 2];
    const float* convw = (const float*)d_in[base + 3];
    const float* convb = (const float*)d_in[base + 4];
    const float* xpw   = (const float*)d_in[base + 5];
    const float* dtw   = (const float*)d_in[base + 6];
    const float* dtb   = (const float*)d_in[base + 7];
    const float* Alog  = (const float*)d_in[base + 8];
    const float* Dp    = (const float*)d_in[base + 9];
    const float* outw  = (const float*)d_in[base + 10];
    const float* fnorm = (const float*)d_in[base + 11];
    int flip = dir;   // dir 1 = backward branch (flipped sequence)

    embed_kernel<<<(unsigned)((embTot + 255) / 256), 256, 0, stream>>>(ids, emb, h, flip);

    for (int l = 0; l < L_; ++l) {
      // xn = rms(h) * norm_l
      rmsnorm_kernel<<<NBS, 256, 0, stream>>>(h, norm + (long)l * DM_, xn,
                                              DM_, DM_, 0, 0);
      // xz = xn @ inw_l^T   (M=4096, N=3072, K=768)
      launch_gemm(xn, DM_, inw + (long)l * 2 * DI_ * DM_, DM_, nullptr,
                  xz, 2 * DI_, NBS, 2 * DI_, DM_, 0, 0, stream);
      // xc = silu(depthwise_conv(xi) + convb)
      conv_silu_kernel<<<(unsigned)((convTot + 255) / 256), 256, 0, stream>>>(
          xz, convw + (long)l * DI_ * KC_, convb + (long)l * DI_, xc);
      // dbc = xc @ xpw_l^T  (M=4096, N=80, K=1536)
      launch_gemm(xc, DI_, xpw + (long)l * DBC_ * DI_, DI_, nullptr,
                  dbc, DBC_, NBS, DBC_, DI_, 0, 0, stream);
      // dt = softplus(dbc[:, :48] @ dtw_l^T + dtb)  (M=4096, N=1536, K=48)
      launch_gemm(dbc, DBC_, dtw + (long)l * DI_ * DR_, DR_, dtb + (long)l * DI_,
                  dt, DI_, NBS, DI_, DR_, 0, 1, stream);
      // sequential selective scan + gate
      scan_kernel<<<B_ * (DI_ / 256), 256, 0, stream>>>(
          dt, dbc, xc, xz, Alog + (long)l * DI_ * DS_, Dp + (long)l * DI_, yb);
      // h += yb @ outw_l^T  (residual, M=4096, N=768, K=1536)
      launch_gemm(yb, DI_, outw + (long)l * DM_ * DI_, DI_, nullptr,
                  h, DM_, NBS, DM_, DI_, 1, 0, stream);
    }

    // final RMSNorm -> concat dest (fw into cols [0,DM), bw flipped into [DM,2DM))
    rmsnorm_kernel<<<NBS, 256, 0, stream>>>(h, fnorm, comb,
                                            DM_, 2L * DM_, flip ? DM_ : 0, flip);
  }

  // logits = comb @ lm_w^T   (M=4096, N=32000, K=1536)
  launch_gemm(comb, 2 * DM_, lm_w, 2 * DM_, nullptr,
              logits, V_, NBS, V_, 2 * DM_, 0, 0, stream);

  // masked mean NLL (single-pass online logsumexp)
  loss_row_kernel<<<NBS, 256, 0, stream>>>(logits, labels, nll);
  loss_reduce_kernel<<<1, 256, 0, stream>>>(nll, logits + (long)NBS * V_);
}